// CombinedCRPSIntervalLoss_34359738621
// MI455X (gfx1250) — compile-verified
//
#include <hip/hip_runtime.h>
#include <hip/hip_bf16.h>

#define S_SAMP 100
#define BLK 128

typedef float v2f __attribute__((ext_vector_type(2)));
typedef float v8f __attribute__((ext_vector_type(8)));

__global__ __launch_bounds__(BLK)
void crps_interval_kernel(const float* __restrict__ mu,
                          const float* __restrict__ sigma,
                          const float* __restrict__ target,
                          const float* __restrict__ noise,
                          float* __restrict__ partial, int N)
{
    __shared__ float smem[S_SAMP * BLK];   // samples, layout [s][tid], bank-conflict free
    __shared__ float red[BLK];

    const int tid = threadIdx.x;
    const int n   = blockIdx.x * BLK + tid;
    const int nc  = (n < N) ? n : (N - 1);          // clamp -> always in-bounds loads

    // ---- Stage noise column into LDS via async global->LDS DMA (ASYNCcnt path) ----
    unsigned ldsa = ((unsigned)(uintptr_t)(&smem[0])) + (unsigned)tid * 4u;
    unsigned goff = (unsigned)nc * 4u;              // byte offset into noise (< 2^31)
    const unsigned gstride = (unsigned)N * 4u;
    #pragma unroll 4
    for (int s = 0; s < S_SAMP; ++s) {
        asm volatile("global_load_async_to_lds_b32 %0, %1, %2"
                     :: "v"(ldsa), "v"(goff), "s"(noise)
                     : "memory");
        ldsa += BLK * 4u;
        goff += gstride;
    }

    const float mu_n   = mu[nc];
    const float sg_raw = sigma[nc];
    const float tg_raw = target[nc];
    const float sigc   = fmaxf(sg_raw, 1e-6f);
    const float tgtc   = fmaxf(tg_raw, 1e-6f);

    // ---- Interval score (closed form, raw sigma/target) ----
    const float ZHI = 1.6448536269514722f;          // norm.ppf(0.95)
    const float lo  = __expf(fmaf(sg_raw, -ZHI, mu_n));
    const float hi  = __expf(fmaf(sg_raw,  ZHI, mu_n));
    const float pen = 20.0f * (fmaxf(lo - tg_raw, 0.0f) + fmaxf(tg_raw - hi, 0.0f));
    const float interval = hi - lo + pen;

    asm volatile("s_wait_asynccnt 0" ::: "memory");

    // ---- Phase A: z -> sample (in place), accumulate term1 ----
    float term1 = 0.0f;
    #pragma unroll 4
    for (int s = 0; s < S_SAMP; ++s) {
        const float z = smem[s * BLK + tid];
        const float v = __expf(fmaf(sigc, z, mu_n));
        term1 += fabsf(v - tgtc);
        smem[s * BLK + tid] = v;
    }

    // ---- Phase B: P = sum_{i<j} |s_i - s_j|, i-blocked by 4 ----
    float P = 0.0f;
    for (int i0 = 0; i0 < S_SAMP; i0 += 4) {
        const float a0 = smem[(i0 + 0) * BLK + tid];
        const float a1 = smem[(i0 + 1) * BLK + tid];
        const float a2 = smem[(i0 + 2) * BLK + tid];
        const float a3 = smem[(i0 + 3) * BLK + tid];
        P += fabsf(a0 - a1) + fabsf(a0 - a2) + fabsf(a0 - a3)
           + fabsf(a1 - a2) + fabsf(a1 - a3) + fabsf(a2 - a3);
        #pragma unroll 4
        for (int j = i0 + 4; j < S_SAMP; ++j) {
            const float b = smem[j * BLK + tid];
            P += fabsf(a0 - b) + fabsf(a1 - b) + fabsf(a2 - b) + fabsf(a3 - b);
        }
    }

    float contrib = 0.0f;
    if (n < N) {
        const float crps = term1 * (1.0f / S_SAMP)
                         - P * (1.0f / ((float)S_SAMP * (float)S_SAMP));
        contrib = crps + interval;
    }
    red[tid] = contrib;
    __syncthreads();

    // ---- Block reduction via V_WMMA_F32_16X16X4_F32 (A = ones => D[m][n] = col-sums) ----
    if (tid < 32) {                                  // wave 0, full EXEC
        v2f A;  A.x = 1.0f; A.y = 1.0f;              // 16x4 all-ones
        v2f B0; B0.x = red[tid];      B0.y = red[tid + 32];
        v2f B1; B1.x = red[tid + 64]; B1.y = red[tid + 96];
        v8f D = {0.f, 0.f, 0.f, 0.f, 0.f, 0.f, 0.f, 0.f};
        D = __builtin_amdgcn_wmma_f32_16x16x4_f32(false, A, false, B0,
                                                  (short)0, D, false, false);
        D = __builtin_amdgcn_wmma_f32_16x16x4_f32(false, A, false, B1,
                                                  (short)0, D, false, false);
        float cs = D[0];                             // lane n (0..15): col-sum n
        cs += __shfl_xor(cs, 1, 32);
        cs += __shfl_xor(cs, 2, 32);
        cs += __shfl_xor(cs, 4, 32);
        cs += __shfl_xor(cs, 8, 32);                 // lanes 0..15 fully summed
        if (tid == 0) partial[blockIdx.x] = cs;
    }
}

__global__ __launch_bounds__(256)
void reduce_kernel(const float* __restrict__ partial, float* __restrict__ out,
                   int nb, float scale)
{
    __shared__ float s[256];
    const int tid = threadIdx.x;
    float a = 0.0f;
    for (int i = tid; i < nb; i += 256) a += partial[i];
    s[tid] = a;
    __syncthreads();
    for (int off = 128; off > 0; off >>= 1) {
        if (tid < off) s[tid] += s[tid + off];
        __syncthreads();
    }
    if (tid == 0) out[0] = s[0] * scale;
}

extern "C" void kernel_launch(void* const* d_in, const int* in_sizes, int n_in,
                              void* d_out, int out_size, void* d_ws, size_t ws_size,
                              hipStream_t stream)
{
    const float* mu     = (const float*)d_in[0];
    const float* sigma  = (const float*)d_in[1];
    const float* target = (const float*)d_in[2];
    const float* noise  = (const float*)d_in[3];   // [S, N] row-major
    const int N = in_sizes[0];
    float* partial = (float*)d_ws;
    float* out = (float*)d_out;

    const int nb = (N + BLK - 1) / BLK;
    crps_interval_kernel<<<nb, BLK, 0, stream>>>(mu, sigma, target, noise, partial, N);
    reduce_kernel<<<1, 256, 0, stream>>>(partial, out, nb, 1.0f / (float)N);
}